// PhysicsMaps_91130616086944
// MI455X (gfx1250) — compile-verified
//
#include <hip/hip_runtime.h>
#include <stdint.h>
#include <math.h>

// ---------------------------------------------------------------------------
// Fused physics-maps stencil for MI455X (gfx1250).
// Bandwidth-bound (235 MB @ 23.3 TB/s ~ 10us floor): strategy is CDNA5 async
// global->LDS staging (ASYNCcnt path) + one-pass LDS tiling, not WMMA.
// ---------------------------------------------------------------------------

#define TS  32          // output tile (TS x TS)
#define GSZ 36          // gray/rgb tile with halo 2
#define GP  37          // padded pitch (odd -> conflict-free columns)
#define DSZ 34          // (gray-mean)^2 tile with halo 1
#define DP  35

__device__ __forceinline__ uint32_t lds_off(const void* p) {
    // flat addr low 32 bits == byte offset within the workgroup LDS allocation
    return (uint32_t)(uintptr_t)p;
}

__device__ __forceinline__ void async_ld_b32(uint32_t ldsaddr, const float* sbase, uint32_t voff) {
    // GLOBAL_LOAD_ASYNC_TO_LDS_B32: vdst = LDS byte address, vaddr = 32-bit
    // unsigned offset, saddr = 64-bit uniform base. Tracked by ASYNCcnt.
    asm volatile("global_load_async_to_lds_b32 %0, %1, %2"
                 :: "v"(ldsaddr), "v"(voff), "s"(sbase)
                 : "memory");
}

__device__ __forceinline__ void wait_async0() {
    asm volatile("s_wait_asynccnt 0x0" ::: "memory");
}

__global__ __launch_bounds__(256) void physics_maps_kernel(const float* __restrict__ x,
                                                           float* __restrict__ out) {
    __shared__ float rgb[3][GSZ][GP];
    __shared__ float gray[GSZ][GP];
    __shared__ float dsq[DSZ][DP];

    const int tid = threadIdx.x;
    const int tx  = blockIdx.x;      // tile col
    const int ty  = blockIdx.y;      // tile row
    const int b   = blockIdx.z;      // batch

    const int H = 512, W = 512;
    const int plane = H * W;                           // 262144 elements
    const float* xb = x + (size_t)b * 3 * plane;       // uniform per block

    const int h0 = ty * TS - 2;
    const int w0 = tx * TS - 2;

    // ---- Stage 1: async-stage the 36x36 RGB halo tile into LDS -------------
    const uint32_t chan_lds = (uint32_t)(GSZ * GP * 4);    // LDS channel stride
    const uint32_t chan_g   = (uint32_t)plane * 4u;        // global channel stride
    for (int i = tid; i < GSZ * GSZ; i += 256) {
        const int r = i / GSZ, c = i % GSZ;
        const int h = h0 + r, w = w0 + c;
        if ((unsigned)h < (unsigned)H && (unsigned)w < (unsigned)W) {
            const uint32_t voff = (uint32_t)((h * W + w) * 4);
            const uint32_t la   = lds_off(&rgb[0][r][c]);
            async_ld_b32(la,                xb, voff);
            async_ld_b32(la + chan_lds,     xb, voff + chan_g);
            async_ld_b32(la + 2 * chan_lds, xb, voff + 2 * chan_g);
        } else {
            // zero padding ('same' conv) for out-of-image texels
            rgb[0][r][c] = 0.f; rgb[1][r][c] = 0.f; rgb[2][r][c] = 0.f;
        }
    }
    wait_async0();
    __syncthreads();

    // ---- Stage 2: grayscale tile ------------------------------------------
    for (int i = tid; i < GSZ * GSZ; i += 256) {
        const int r = i / GSZ, c = i % GSZ;
        gray[r][c] = 0.299f * rgb[0][r][c] + 0.587f * rgb[1][r][c]
                   + 0.114f * rgb[2][r][c];
    }
    __syncthreads();

    // ---- Stage 3: (gray - local_mean)^2 on the 34x34 halo-1 tile ----------
    // Outside the image, the *product* is padded with zero (XLA pads the conv
    // input, not the mean), so force 0 there.
    for (int i = tid; i < DSZ * DSZ; i += 256) {
        const int r = i / DSZ, c = i % DSZ;
        const int gr = r + 1, gc = c + 1;    // gray index of this point
        float s = 0.f;
        #pragma unroll
        for (int dy = -1; dy <= 1; ++dy)
            #pragma unroll
            for (int dx = -1; dx <= 1; ++dx)
                s += gray[gr + dy][gc + dx];
        const float m   = s * (1.f / 9.f);
        const float dev = gray[gr][gc] - m;
        const int h = h0 + 1 + r, w = w0 + 1 + c;     // global coords
        const bool inside = ((unsigned)h < (unsigned)H) && ((unsigned)w < (unsigned)W);
        dsq[r][c] = inside ? dev * dev : 0.f;
    }
    __syncthreads();

    // ---- Stage 4: 4 output pixels per thread, 4 coalesced plane stores ----
    float* ob = out + (size_t)b * 4 * plane;
    #pragma unroll
    for (int k = 0; k < 4; ++k) {
        const int idx = tid + k * 256;
        const int pr = idx >> 5, pc = idx & 31;
        const int gr = pr + 2, gc = pc + 2;

        const float g00 = gray[gr-1][gc-1], g01 = gray[gr-1][gc], g02 = gray[gr-1][gc+1];
        const float g10 = gray[gr  ][gc-1], g11 = gray[gr  ][gc], g12 = gray[gr  ][gc+1];
        const float g20 = gray[gr+1][gc-1], g21 = gray[gr+1][gc], g22 = gray[gr+1][gc+1];

        // Sobel (all consumers are sign-invariant: sqrt(gx^2+gy^2), |gx|+|gy|)
        const float gx  = (g02 - g00) + 2.f * (g12 - g10) + (g22 - g20);
        const float gy  = (g20 - g00) + 2.f * (g21 - g01) + (g22 - g02);
        const float lap = g01 + g10 + g12 + g21 - 4.f * g11;

        const float grad  = sqrtf(gx * gx + gy * gy + 1e-8f);
        const float var   = (dsq[pr  ][pc] + dsq[pr  ][pc+1] + dsq[pr  ][pc+2]
                           + dsq[pr+1][pc] + dsq[pr+1][pc+1] + dsq[pr+1][pc+2]
                           + dsq[pr+2][pc] + dsq[pr+2][pc+1] + dsq[pr+2][pc+2]) * (1.f / 9.f);
        const float noise = fabsf(lap);
        const float edge  = 0.5f * (fabsf(gx) + fabsf(gy));

        const int h = ty * TS + pr, w = tx * TS + pc;
        const size_t o = (size_t)(h * W + w);
        ob[o]                      = grad;
        ob[o + (size_t)plane]      = var;
        ob[o + 2 * (size_t)plane]  = noise;
        ob[o + 3 * (size_t)plane]  = edge;
    }
}

extern "C" void kernel_launch(void* const* d_in, const int* in_sizes, int n_in,
                              void* d_out, int out_size, void* d_ws, size_t ws_size,
                              hipStream_t stream) {
    const float* x = (const float*)d_in[0];
    float* out = (float*)d_out;
    (void)in_sizes; (void)n_in; (void)out_size; (void)d_ws; (void)ws_size;

    dim3 grid(512 / TS, 512 / TS, 32);   // 16 x 16 tiles x 32 batches = 8192 blocks
    physics_maps_kernel<<<grid, 256, 0, stream>>>(x, out);
}